// NARLD_79044578115695
// MI455X (gfx1250) — compile-verified
//
#include <hip/hip_runtime.h>

// MI455X (gfx1250) Richardson-Lucy deconvolution.
// FFT2 as DFT-matmul (F·X·F) on V_WMMA_F32_16X16X4_F32, with Tensor Data
// Mover (tensor_load_to_lds) double-buffered global->LDS panel staging.

typedef __attribute__((ext_vector_type(2))) float        v2f;
typedef __attribute__((ext_vector_type(8))) float        v8f;
typedef __attribute__((ext_vector_type(4))) unsigned int v4u;
typedef __attribute__((ext_vector_type(4))) int          v4i;
typedef __attribute__((ext_vector_type(8))) int          v8i;

constexpr int   kNXY    = 512;
constexpr int   kEXT    = 64;
constexpr int   kNZ     = 100;
constexpr int   kROI    = 128;
constexpr int   kPXY    = 384;            // kNXY - 2*kEXT
constexpr int   kN2     = kNXY * kNXY;    // 262144
constexpr int   kC0     = kNXY / 2 - kROI;      // 128
constexpr int   kOBJ    = 2 * kROI;             // 256
constexpr int   kOBJ2   = kOBJ * kOBJ;          // 65536
constexpr float kBKG    = 110.0f;
constexpr float kSNR    = 200.0f;

// ---------------------------------------------------------------------------
// TDM: DMA one 2-D tile (rows x tileW f32 elements, row stride = strideElems)
// from global memory into LDS at byte offset ldsOff. Optional LDS padding of
// 1 DWORD every 32 DWORDs (reproduces a [rows][33] conflict-free layout).
// Descriptor per CDNA5 ISA section 8.3/8.4 (2-D tensor: groups 2/3 zero).
// ---------------------------------------------------------------------------
static __device__ __forceinline__ void tdm_load_2d(unsigned ldsOff, const float* gptr,
                                                   int tileW, int rows, int strideElems,
                                                   int padded) {
    unsigned long long ga = (unsigned long long)gptr;
    v4u g0;
    g0[0] = 1u;                                            // count=1, user desc
    g0[1] = ldsOff;                                        // lds_addr (bytes)
    g0[2] = (unsigned)(ga & 0xFFFFFFFFu);                  // global_addr[31:0]
    g0[3] = (unsigned)((ga >> 32) & 0x1FFFFFFu) | (2u << 30); // [56:32] | type=2
    unsigned d0 = (2u << 16);                              // data_size = 4 bytes
    if (padded) d0 |= (1u << 20) | (4u << 22);             // pad_en, 32 DW interval, 1 DW pad
    v8i g1;
    g1[0] = (int)d0;                                       // wg_mask=0 (not in cluster)
    g1[1] = (int)(((unsigned)strideElems & 0xFFFFu) << 16);    // tensor_dim0[15:0]
    g1[2] = (int)((((unsigned)strideElems >> 16) & 0xFFFFu) | (512u << 16)); // dim0 hi | dim1 lo
    g1[3] = (int)(((unsigned)tileW & 0xFFFFu) << 16);      // dim1 hi=0 | tile_dim0
    g1[4] = rows;                                          // tile_dim1 (tile_dim2=0)
    g1[5] = strideElems;                                   // tensor_dim0_stride[31:0]
    g1[6] = 0;                                             // stride hi | dim1_stride lo
    g1[7] = 0;
    v4i z4 = {0, 0, 0, 0};
#if defined(__clang_major__) && (__clang_major__ >= 23)
    v8i z8 = {0, 0, 0, 0, 0, 0, 0, 0};
    __builtin_amdgcn_tensor_load_to_lds(g0, g1, z4, z4, z8, 0);
#else
    __builtin_amdgcn_tensor_load_to_lds(g0, g1, z4, z4, 0);
#endif
}

#define WMMA_F32(A, B, C) \
    __builtin_amdgcn_wmma_f32_16x16x4_f32(false, (A), false, (B), (short)0, (C), false, false)

// ---------------------------------------------------------------------------
// Complex GEMM, fixed 512x512x512, batched over blockIdx.z.
// C = alpha * A * B  (complex, split real/imag planes, row-major, ld = 512)
// Block: 128 threads = 4 wave32 waves. Block tile 32x64; each wave owns a
// 16x32 strip (two 16x16 WMMA tiles sharing the A fragments). K is staged in
// 32-wide panels by the Tensor Data Mover, double-buffered: wave 0 issues the
// next panel's 4 TDM loads, s_wait_tensorcnt's the previous, then a barrier
// releases the compute waves.
// ---------------------------------------------------------------------------
__global__ __launch_bounds__(128) void cgemm512(
    const float* __restrict__ Ar, const float* __restrict__ Ai, long batA,
    const float* __restrict__ Br, const float* __restrict__ Bi, long batB,
    float* __restrict__ Cr, float* __restrict__ Ci, long batC,
    float alpha)
{
    __shared__ float AsR[2][32][33], AsI[2][32][33];
    __shared__ float BsR[2][32][64], BsI[2][32][64];

    const int z = blockIdx.z;
    Ar += (long)z * batA;  Ai += (long)z * batA;
    Br += (long)z * batB;  Bi += (long)z * batB;
    Cr += (long)z * batC;  Ci += (long)z * batC;

    const int t    = threadIdx.x;
    const int ty   = t >> 5;           // wave id 0..3
    const int lane = t & 31;
    const int m0   = blockIdx.y * 32;
    const int n0   = blockIdx.x * 64;
    const int wm   = (ty >> 1) * 16;   // wave row sub-tile
    const int wn   = (ty & 1) * 32;    // wave col strip (two 16-wide tiles)
    const int half = lane >> 4;        // ISA: lanes 16..31 carry K+2 / M+8
    const int mr   = lane & 15;

    v8f aRR0 = {0.f,0.f,0.f,0.f,0.f,0.f,0.f,0.f}, aII0 = aRR0, aCI0 = aRR0;
    v8f aRR1 = aRR0, aII1 = aRR0, aCI1 = aRR0;

    // Generic->LDS address: low 32 bits of the flat address are the LDS offset
    // (shared aperture keeps the offset in addr[31:0]).
    const unsigned offAR0 = (unsigned)(unsigned long long)&AsR[0][0][0];
    const unsigned offAI0 = (unsigned)(unsigned long long)&AsI[0][0][0];
    const unsigned offBR0 = (unsigned)(unsigned long long)&BsR[0][0][0];
    const unsigned offBI0 = (unsigned)(unsigned long long)&BsI[0][0][0];
    const unsigned szA = 32 * 33 * 4, szB = 32 * 64 * 4;

    if (ty == 0) {                     // prologue: panel 0 into buffer 0
        tdm_load_2d(offAR0, Ar + (long)m0 * 512, 32, 32, 512, 1);
        tdm_load_2d(offAI0, Ai + (long)m0 * 512, 32, 32, 512, 1);
        tdm_load_2d(offBR0, Br + n0, 64, 32, 512, 0);
        tdm_load_2d(offBI0, Bi + n0, 64, 32, 512, 0);
    }

    for (int i = 0; i < 16; ++i) {
        const int buf = i & 1;
        if (ty == 0) {
            if (i + 1 < 16) {
                const int nb = buf ^ 1;
                const long kc = (long)(i + 1) * 32;
                tdm_load_2d(offAR0 + nb * szA, Ar + (long)m0 * 512 + kc, 32, 32, 512, 1);
                tdm_load_2d(offAI0 + nb * szA, Ai + (long)m0 * 512 + kc, 32, 32, 512, 1);
                tdm_load_2d(offBR0 + nb * szB, Br + kc * 512 + n0, 64, 32, 512, 0);
                tdm_load_2d(offBI0 + nb * szB, Bi + kc * 512 + n0, 64, 32, 512, 0);
                __builtin_amdgcn_s_wait_tensorcnt(4);  // panel i landed
            } else {
                __builtin_amdgcn_s_wait_tensorcnt(0);
            }
        }
        __syncthreads();

#pragma unroll
        for (int kk = 0; kk < 32; kk += 4) {
            const int ka = kk + 2 * half;
            v2f aR  = { AsR[buf][wm + mr][ka],      AsR[buf][wm + mr][ka + 1] };
            v2f aI  = { AsI[buf][wm + mr][ka],      AsI[buf][wm + mr][ka + 1] };
            v2f bR0 = { BsR[buf][ka][wn + mr],      BsR[buf][ka + 1][wn + mr] };
            v2f bI0 = { BsI[buf][ka][wn + mr],      BsI[buf][ka + 1][wn + mr] };
            v2f bR1 = { BsR[buf][ka][wn + 16 + mr], BsR[buf][ka + 1][wn + 16 + mr] };
            v2f bI1 = { BsI[buf][ka][wn + 16 + mr], BsI[buf][ka + 1][wn + 16 + mr] };
            aRR0 = WMMA_F32(aR, bR0, aRR0);
            aII0 = WMMA_F32(aI, bI0, aII0);
            aCI0 = WMMA_F32(aR, bI0, aCI0);
            aCI0 = WMMA_F32(aI, bR0, aCI0);
            aRR1 = WMMA_F32(aR, bR1, aRR1);
            aII1 = WMMA_F32(aI, bI1, aII1);
            aCI1 = WMMA_F32(aR, bI1, aCI1);
            aCI1 = WMMA_F32(aI, bR1, aCI1);
        }
        __syncthreads();
    }

    // C/D layout: VGPR v -> row (v + 8*half), lanes 0..15/16..31 -> col (lane&15)
    const int gm  = m0 + wm + 8 * half;
    const int gn0 = n0 + wn + mr;
#pragma unroll
    for (int v = 0; v < 8; ++v) {
        Cr[(long)(gm + v) * 512 + gn0]      = alpha * (aRR0[v] - aII0[v]);
        Ci[(long)(gm + v) * 512 + gn0]      = alpha * aCI0[v];
        Cr[(long)(gm + v) * 512 + gn0 + 16] = alpha * (aRR1[v] - aII1[v]);
        Ci[(long)(gm + v) * 512 + gn0 + 16] = alpha * aCI1[v];
    }
}

// ---------------------------------------------------------------------------
// Setup / pointwise kernels
// ---------------------------------------------------------------------------
__global__ void build_dft(float* Fr, float* Fi, float* Gr, float* Gi) {
    int p = blockIdx.x * blockDim.x + threadIdx.x;
    if (p >= kN2) return;
    int i = p >> 9, j = p & 511;
    int m = (i * j) & 511;                       // exact modular phase
    float ang = -6.283185307179586f * (float)m / 512.0f;
    float s, c;
    __sincosf(ang, &s, &c);
    Fr[p] = c; Fi[p] = s;                        // F (forward)
    Gr[p] = c; Gi[p] = -s;                       // conj(F) (inverse)
}

__global__ void psf_sum_k(const float* __restrict__ PSF, float* __restrict__ recip) {
    __shared__ float sh[256];
    const int z = blockIdx.x;
    float s = 0.f;
    for (int i = threadIdx.x; i < kPXY * kPXY; i += 256)
        s += PSF[(long)i * kNZ + z];             // PSF[y][x][z]
    sh[threadIdx.x] = s;
    __syncthreads();
    for (int o = 128; o > 0; o >>= 1) {
        if (threadIdx.x < o) sh[threadIdx.x] += sh[threadIdx.x + o];
        __syncthreads();
    }
    if (threadIdx.x == 0) recip[z] = 1.0f / sh[0];
}

// P padded + ifftshift (roll by N/2 on even N), imag plane zeroed.
__global__ void pad_shift(const float* __restrict__ PSF, float* __restrict__ oR, float* __restrict__ oI) {
    long i = (long)blockIdx.x * blockDim.x + threadIdx.x;
    if (i >= (long)kNZ * kN2) return;
    int z = (int)(i / kN2), rem = (int)(i % kN2);
    int y = rem >> 9, x = rem & 511;
    int ys = (y + 256) & 511, xs = (x + 256) & 511;
    int yy = ys - kEXT, xx = xs - kEXT;
    float v = 0.f;
    if (yy >= 0 && yy < kPXY && xx >= 0 && xx < kPXY)
        v = PSF[((long)yy * kPXY + xx) * kNZ + z];
    oR[i] = v; oI[i] = 0.f;
}

__global__ void img_clip(const float* __restrict__ src, float* __restrict__ dst) {
    int p = blockIdx.x * blockDim.x + threadIdx.x;
    if (p < kPXY * kPXY) dst[p] = fmaxf(src[p] - kBKG, 0.f);
}

__global__ void embed_obj(const float* __restrict__ obj, float* __restrict__ oR, float* __restrict__ oI) {
    long i = (long)blockIdx.x * blockDim.x + threadIdx.x;
    if (i >= (long)kNZ * kN2) return;
    int z = (int)(i / kN2), rem = (int)(i % kN2);
    int y = rem >> 9, x = rem & 511;
    float v = 0.f;
    if (y >= kC0 && y < kC0 + kOBJ && x >= kC0 && x < kC0 + kOBJ)
        v = obj[(long)z * kOBJ2 + (y - kC0) * kOBJ + (x - kC0)];
    oR[i] = v; oI[i] = 0.f;
}

__global__ void cmul_inplace(float* ar, float* ai,
                             const float* __restrict__ br, const float* __restrict__ bi, long n) {
    long i = (long)blockIdx.x * blockDim.x + threadIdx.x;
    if (i >= n) return;
    float x = ar[i], y = ai[i], c = br[i], d = bi[i];
    ar[i] = x * c - y * d;
    ai[i] = x * d + y * c;
}

// out[z] = fftR * conj(fftP[z])   (fftR broadcast over z)
__global__ void cmul_conj_bcast(const float* __restrict__ xr, const float* __restrict__ xi,
                                const float* __restrict__ pr, const float* __restrict__ pi,
                                float* __restrict__ orr, float* __restrict__ oii, long n) {
    long i = (long)blockIdx.x * blockDim.x + threadIdx.x;
    if (i >= n) return;
    long p = i % kN2;
    float x = xr[p], y = xi[p], c = pr[i], d = pi[i];
    orr[i] = x * c + y * d;
    oii[i] = y * c - x * d;
}

// ImgEst[p] = sum_z max(est_re[z][p] / psf_sum[z], 0)
__global__ void img_est(const float* __restrict__ estR, const float* __restrict__ recip,
                        float* __restrict__ imgEst) {
    int p = blockIdx.x * blockDim.x + threadIdx.x;
    if (p >= kN2) return;
    float s = 0.f;
    for (int z = 0; z < kNZ; ++z)
        s += fmaxf(estR[(long)z * kN2 + p] * recip[z], 0.f);
    imgEst[p] = s;
}

// ---- exact median via 3-pass radix select on float bit patterns (>=0) -----
__global__ void sel_init(unsigned* st, unsigned rank) { st[0] = 0u; st[1] = 0u; st[2] = rank; }

__global__ void hist_zero(unsigned* hist) { hist[blockIdx.x * blockDim.x + threadIdx.x] = 0u; }

__global__ void sel_hist(const float* __restrict__ v, int n, const unsigned* __restrict__ st,
                         unsigned* __restrict__ hist, int shift, unsigned binMask) {
    __shared__ unsigned h[2048];
    for (int i = threadIdx.x; i < 2048; i += blockDim.x) h[i] = 0u;
    __syncthreads();
    const unsigned prefix = st[0], mask = st[1];
    for (int i = blockIdx.x * blockDim.x + threadIdx.x; i < n; i += gridDim.x * blockDim.x) {
        unsigned u = __float_as_uint(v[i]);
        if ((u & mask) == prefix) atomicAdd(&h[(u >> shift) & binMask], 1u);
    }
    __syncthreads();
    for (int i = threadIdx.x; i <= (int)binMask; i += blockDim.x)
        if (h[i]) atomicAdd(&hist[i], h[i]);
}

__global__ void sel_pick(unsigned* st, const unsigned* __restrict__ hist,
                         int shift, int bins, int isFinal, float* outMed) {
    unsigned r = st[2], cum = 0;
    int b = 0;
    for (; b < bins - 1; ++b) {
        unsigned c = hist[b];
        if (cum + c > r) break;
        cum += c;
    }
    st[0] |= ((unsigned)b) << shift;
    st[1] |= ((unsigned)(bins - 1)) << shift;
    st[2] = r - cum;
    if (isFinal) *outMed = __uint_as_float(st[0]);
}

__global__ void tmp_final(const float* __restrict__ med, float* __restrict__ tscale) {
    tscale[0] = 0.5f * (med[0] + med[1]) / kSNR;
}

__global__ void build_ratio(const float* __restrict__ img, const float* __restrict__ imgEst,
                            const float* __restrict__ tscale,
                            float* __restrict__ Rr, float* __restrict__ Ri) {
    int p = blockIdx.x * blockDim.x + threadIdx.x;
    if (p >= kN2) return;
    int y = p >> 9, x = p & 511;
    float r = 1.f;
    if (y >= kEXT && y < kNXY - kEXT && x >= kEXT && x < kNXY - kEXT)
        r = img[(y - kEXT) * kPXY + (x - kEXT)] / (imgEst[p] + tscale[0]);
    Rr[p] = r; Ri[p] = 0.f;
}

// obj_out = obj_in * max(back_re[z][y+c0][x+c0] / psf_sum[z], 0)
__global__ void update_obj(const float* __restrict__ src, const float* __restrict__ backR,
                           const float* __restrict__ recip, float* __restrict__ dst) {
    long i = (long)blockIdx.x * blockDim.x + threadIdx.x;
    if (i >= (long)kNZ * kOBJ2) return;
    int z = (int)(i / kOBJ2), rem = (int)(i % kOBJ2);
    int y = rem >> 8, x = rem & 255;
    float b = backR[(long)z * kN2 + (long)(y + kC0) * kNXY + (x + kC0)] * recip[z];
    dst[i] = src[i] * fmaxf(b, 0.f);
}

// ---------------------------------------------------------------------------
extern "C" void kernel_launch(void* const* d_in, const int* in_sizes, int n_in,
                              void* d_out, int out_size, void* d_ws, size_t ws_size,
                              hipStream_t stream) {
    (void)in_sizes; (void)n_in; (void)out_size; (void)ws_size;
    const float* imstack = (const float*)d_in[0];   // 384*384
    const float* init    = (const float*)d_in[1];   // 100*256*256
    const float* PSF     = (const float*)d_in[2];   // 384*384*100
    float* out = (float*)d_out;                     // 100*256*256 f32

    char* cur = (char*)d_ws;
    auto alloc = [&](size_t nf) -> float* {
        float* p = (float*)cur;
        cur += ((nf * 4 + 255) & ~(size_t)255);
        return p;
    };
    const size_t ZN2 = (size_t)kNZ * kN2;
    float *Fr = alloc(kN2), *Fi = alloc(kN2), *Gr = alloc(kN2), *Gi = alloc(kN2);
    float *fPr = alloc(ZN2), *fPi = alloc(ZN2);
    float *Ar = alloc(ZN2), *Ai = alloc(ZN2);
    float *Br = alloc(ZN2), *Bi = alloc(ZN2);
    float *objbuf = alloc((size_t)kNZ * kOBJ2);
    float *recip  = alloc(kNZ);
    float *img    = alloc(kPXY * kPXY);
    float *imgEst = alloc(kN2);
    float *Rr = alloc(kN2), *Ri = alloc(kN2), *FRr = alloc(kN2), *FRi = alloc(kN2);
    float *med = alloc(2), *tscale = alloc(1);
    unsigned* st   = (unsigned*)alloc(4);
    unsigned* hist = (unsigned*)alloc(2048);

    const dim3 gB(128), gemmB(8, 16, kNZ), gemm1(8, 16, 1);
    const int PW = (kPXY * kPXY + 255) / 256;
    const int NB = (kN2 + 255) / 256;
    const int ZB = (int)((ZN2 + 255) / 256);
    const int OB = (kNZ * kOBJ2 + 255) / 256;
    const long LN2 = (long)kN2;
    const float invN2 = 1.0f / (float)kN2;

    // --- one-time precompute ---
    build_dft<<<NB, 256, 0, stream>>>(Fr, Fi, Gr, Gi);
    psf_sum_k<<<kNZ, 256, 0, stream>>>(PSF, recip);
    pad_shift<<<ZB, 256, 0, stream>>>(PSF, Ar, Ai);
    cgemm512<<<gemmB, gB, 0, stream>>>(Fr, Fi, 0L, Ar, Ai, LN2, Br, Bi, LN2, 1.f);   // F * P
    cgemm512<<<gemmB, gB, 0, stream>>>(Br, Bi, LN2, Fr, Fi, 0L, fPr, fPi, LN2, 1.f); // (F*P) * F
    img_clip<<<PW, 256, 0, stream>>>(imstack, img);

    for (int it = 0; it < 2; ++it) {
        const float* objSrc = (it == 0) ? init : objbuf;
        float* objDst = (it == 1) ? out : objbuf;

        // est = ifft2(fft2(pad(obj)) * fftP)
        embed_obj<<<ZB, 256, 0, stream>>>(objSrc, Ar, Ai);
        cgemm512<<<gemmB, gB, 0, stream>>>(Fr, Fi, 0L, Ar, Ai, LN2, Br, Bi, LN2, 1.f);
        cgemm512<<<gemmB, gB, 0, stream>>>(Br, Bi, LN2, Fr, Fi, 0L, Ar, Ai, LN2, 1.f);
        cmul_inplace<<<ZB, 256, 0, stream>>>(Ar, Ai, fPr, fPi, (long)ZN2);
        cgemm512<<<gemmB, gB, 0, stream>>>(Gr, Gi, 0L, Ar, Ai, LN2, Br, Bi, LN2, 1.f);
        cgemm512<<<gemmB, gB, 0, stream>>>(Br, Bi, LN2, Gr, Gi, 0L, Ar, Ai, LN2, invN2);

        // ImgEst, exact median of both middle ranks, Ratio
        img_est<<<NB, 256, 0, stream>>>(Ar, recip, imgEst);
        for (int k = 0; k < 2; ++k) {
            sel_init<<<1, 1, 0, stream>>>(st, 131071u + (unsigned)k);
            hist_zero<<<8, 256, 0, stream>>>(hist);
            sel_hist<<<256, 256, 0, stream>>>(imgEst, kN2, st, hist, 21, 2047u);
            sel_pick<<<1, 1, 0, stream>>>(st, hist, 21, 2048, 0, med + k);
            hist_zero<<<8, 256, 0, stream>>>(hist);
            sel_hist<<<256, 256, 0, stream>>>(imgEst, kN2, st, hist, 10, 2047u);
            sel_pick<<<1, 1, 0, stream>>>(st, hist, 10, 2048, 0, med + k);
            hist_zero<<<8, 256, 0, stream>>>(hist);
            sel_hist<<<256, 256, 0, stream>>>(imgEst, kN2, st, hist, 0, 1023u);
            sel_pick<<<1, 1, 0, stream>>>(st, hist, 0, 1024, 1, med + k);
        }
        tmp_final<<<1, 1, 0, stream>>>(med, tscale);
        build_ratio<<<NB, 256, 0, stream>>>(img, imgEst, tscale, Rr, Ri);

        // back = ifft2(fft2(Ratio) * conj(fftP)); obj *= max(back,0) window
        cgemm512<<<gemm1, gB, 0, stream>>>(Fr, Fi, 0L, Rr, Ri, 0L, Ar, Ai, 0L, 1.f);
        cgemm512<<<gemm1, gB, 0, stream>>>(Ar, Ai, 0L, Fr, Fi, 0L, FRr, FRi, 0L, 1.f);
        cmul_conj_bcast<<<ZB, 256, 0, stream>>>(FRr, FRi, fPr, fPi, Br, Bi, (long)ZN2);
        cgemm512<<<gemmB, gB, 0, stream>>>(Gr, Gi, 0L, Br, Bi, LN2, Ar, Ai, LN2, 1.f);
        cgemm512<<<gemmB, gB, 0, stream>>>(Ar, Ai, LN2, Gr, Gi, 0L, Br, Bi, LN2, invN2);
        update_obj<<<OB, 256, 0, stream>>>(objSrc, Br, recip, objDst);
    }
}